// MoSca_39591008534749
// MI455X (gfx1250) — compile-verified
//
#include <hip/hip_runtime.h>

// MoSca robust topology distance + KNN for gfx1250 (MI455X, wave32).
// Per frame per 16x16 tile, two V_WMMA_F32_16X16X4_F32:
//   c1 = A x B + C:  A[i,:] = [-2x_i, -2y_i, -2z_i, |x_i|^2], B[:,j] = [x_j,y_j,z_j,1],
//                    C = broadcast(|x_j|^2)  =>  c1[i,j] = d2(i,j)  (full rank-5 form)
//   c2 = mask outer product m_i * m_j (per-pair mask + valid-frame count)
// Top-8-over-time kept per pair in registers on masked d^2 (sqrt is monotonic,
// so percentile selection commutes with it; single v_sqrt_f32 at tile end).

typedef __attribute__((ext_vector_type(2))) float v2f;
typedef __attribute__((ext_vector_type(8))) float v8f;

#define T_FRAMES 80
#define N_PTS   1024
#define TOPK    8
#define KNN     16
#define INVALID_DIST 1e10f

// ---------------------------------------------------------------------------
// Kernel 1: pack [x,y,z,|x|^2] and mask->float
// ---------------------------------------------------------------------------
__global__ void mosca_pack_kernel(const float* __restrict__ xyz,
                                  const int* __restrict__ mask,
                                  float4* __restrict__ p4,
                                  float* __restrict__ mf,
                                  int tn) {
    int idx = blockIdx.x * blockDim.x + threadIdx.x;
    if (idx >= tn) return;
    float x = xyz[3 * idx + 0];
    float y = xyz[3 * idx + 1];
    float z = xyz[3 * idx + 2];
    float4 p; p.x = x; p.y = y; p.z = z; p.w = x * x + y * y + z * z;
    p4[idx] = p;
    mf[idx] = mask[idx] ? 1.0f : 0.0f;
}

// ---------------------------------------------------------------------------
// Kernel 2: one wave (32 lanes) per 16x16 (i,j) tile.
// ---------------------------------------------------------------------------
__global__ __launch_bounds__(32)
void mosca_tile_kernel(const float4* __restrict__ p4,
                       const float* __restrict__ mf,
                       float* __restrict__ D,
                       int T, int N) {
    const int lane = threadIdx.x;        // 0..31
    const int half = lane >> 4;          // 0: feeds K0/K1, 1: feeds K2/K3
    const int lj   = lane & 15;          // row M (A) / col N (B,C)
    const int i0   = blockIdx.y * 16;
    const int j0   = blockIdx.x * 16;
    const int ri   = i0 + lj;            // A-side point this lane feeds
    const int cj   = j0 + lj;            // B-side point this lane feeds

    float top[8][TOPK];                  // per C-vgpr descending top-8 of d^2
    float cnt[8];                        // valid-frame counts (exact ints <= 80)
#pragma unroll
    for (int v = 0; v < 8; ++v) {
        cnt[v] = 0.0f;
#pragma unroll
        for (int s = 0; s < TOPK; ++s) top[v][s] = 0.0f;
    }

    for (int t = 0; t < T; ++t) {
        const int base = t * N;
        float4 rp = p4[base + ri];
        float4 cp = p4[base + cj];
        float  mi = mf[base + ri];
        float  mj = mf[base + cj];

        v2f a, b, am, bm;
        // A row ri: K0=-2x K1=-2y (lanes 0-15), K2=-2z K3=sq (lanes 16-31)
        a.x = half ? (-2.0f * rp.z) : (-2.0f * rp.x);
        a.y = half ? rp.w           : (-2.0f * rp.y);
        // B col cj: K0=x K1=y (lanes 0-15), K2=z K3=1 (lanes 16-31)
        b.x = half ? cp.z : cp.x;
        b.y = half ? 1.0f : cp.y;
        // Mask outer product through K=0 only.
        am.x = half ? 0.0f : mi;  am.y = 0.0f;
        bm.x = half ? 0.0f : mj;  bm.y = 0.0f;

        // C = broadcast(sq_j): WMMA completes d2 = sq_i + sq_j - 2*dot in one op.
        const float sqj = cp.w;
        v8f c = {sqj, sqj, sqj, sqj, sqj, sqj, sqj, sqj};
        c = __builtin_amdgcn_wmma_f32_16x16x4_f32(false, a, false, b,
                                                  (short)0, c, false, false);
        v8f m = {};
        m = __builtin_amdgcn_wmma_f32_16x16x4_f32(false, am, false, bm,
                                                  (short)0, m, false, false);
#pragma unroll
        for (int v = 0; v < 8; ++v) {
            float d2 = m[v] * fmaxf(c[v], 0.0f); // masked-out frames -> 0
            cnt[v] += m[v];
            float tv = d2;                        // branchless descending insert
#pragma unroll
            for (int s = 0; s < TOPK; ++s) {
                float hi = fmaxf(top[v][s], tv);
                float lo = fminf(top[v][s], tv);
                top[v][s] = hi;
                tv = lo;
            }
        }
    }

#pragma unroll
    for (int v = 0; v < 8; ++v) {
        int mc  = (int)(cnt[v] + 0.5f);
        int sel = (TOPK * mc + (T - 1)) / T - 1;   // ceil(top_k*cnt/T)-1
        sel = sel < 0 ? 0 : (sel > TOPK - 1 ? TOPK - 1 : sel);
        float v2 = top[v][0];
#pragma unroll
        for (int s = 1; s < TOPK; ++s) v2 = (sel == s) ? top[v][s] : v2;
        float val = __builtin_amdgcn_sqrtf(v2);    // single v_sqrt_f32
        const int i = i0 + v + 8 * half;           // C layout: vgpr v -> row v / v+8
        const int j = j0 + lj;
        // zero distance or diagonal -> invalid (matches reference `_d==0 -> 1e10`)
        val = (v2 == 0.0f || i == j) ? INVALID_DIST : val;
        D[(size_t)i * N + j] = val;
    }
}

// ---------------------------------------------------------------------------
// Kernel 3: one wave per row -> K=16 smallest with lax.top_k tie semantics
// (sorted by (dist asc, index asc)).
// ---------------------------------------------------------------------------
__global__ __launch_bounds__(32)
void mosca_knn_kernel(const float* __restrict__ D,
                      float* __restrict__ od,
                      int* __restrict__ oi,
                      int N) {
    __shared__ float sd[32 * KNN];
    __shared__ int   si[32 * KNN];
    const int row  = blockIdx.x;
    const int lane = threadIdx.x;

    float kd[KNN]; int ki[KNN];
#pragma unroll
    for (int s = 0; s < KNN; ++s) { kd[s] = 3.4e38f; ki[s] = 0x7fffffff; }

    for (int j = lane; j < N; j += 32) {
        float v  = D[(size_t)row * N + j];
        int   vi = j;
#pragma unroll
        for (int s = 0; s < KNN; ++s) {
            bool lt = (v < kd[s]) || ((v == kd[s]) && (vi < ki[s]));
            float nd = lt ? v     : kd[s];
            int   ni = lt ? vi    : ki[s];
            float xd = lt ? kd[s] : v;
            int   xi = lt ? ki[s] : vi;
            kd[s] = nd; ki[s] = ni; v = xd; vi = xi;
        }
    }
#pragma unroll
    for (int s = 0; s < KNN; ++s) { sd[lane * KNN + s] = kd[s]; si[lane * KNN + s] = ki[s]; }
    __syncthreads();

    if (lane == 0) {
        float fd[KNN]; int fi[KNN];
#pragma unroll
        for (int s = 0; s < KNN; ++s) { fd[s] = 3.4e38f; fi[s] = 0x7fffffff; }
        for (int q = 0; q < 32 * KNN; ++q) {
            float v  = sd[q];
            int   vi = si[q];
#pragma unroll
            for (int s = 0; s < KNN; ++s) {
                bool lt = (v < fd[s]) || ((v == fd[s]) && (vi < fi[s]));
                float nd = lt ? v     : fd[s];
                int   ni = lt ? vi    : fi[s];
                float xd = lt ? fd[s] : v;
                int   xi = lt ? fi[s] : vi;
                fd[s] = nd; fi[s] = ni; v = xd; vi = xi;
            }
        }
#pragma unroll
        for (int s = 0; s < KNN; ++s) {
            od[(size_t)row * KNN + s] = fd[s];
            oi[(size_t)row * KNN + s] = fi[s];
        }
    }
}

// ---------------------------------------------------------------------------
extern "C" void kernel_launch(void* const* d_in, const int* in_sizes, int n_in,
                              void* d_out, int out_size, void* d_ws, size_t ws_size,
                              hipStream_t stream) {
    (void)in_sizes; (void)n_in; (void)out_size; (void)ws_size;
    const int T = T_FRAMES, N = N_PTS;

    const float* xyz  = (const float*)d_in[0];   // [T,N,3] f32
    const int*   mask = (const int*)d_in[1];     // [T,N] bool->int
    // d_in[2]=top_k(8), d_in[3]=K(16): compiled-in constants.

    float4* p4 = (float4*)d_ws;                                   // T*N float4
    float*  mf = (float*)((char*)d_ws + (size_t)T * N * sizeof(float4));

    float* Dm   = (float*)d_out;                 // [N,N]
    float* knnd = Dm + (size_t)N * N;            // [N,KNN] f32
    int*   knni = (int*)(knnd + (size_t)N * KNN);// [N,KNN] i32 (raw bits)

    const int tn = T * N;
    mosca_pack_kernel<<<(tn + 255) / 256, 256, 0, stream>>>(xyz, mask, p4, mf, tn);

    dim3 grid(N / 16, N / 16);
    mosca_tile_kernel<<<grid, 32, 0, stream>>>(p4, mf, Dm, T, N);

    mosca_knn_kernel<<<N, 32, 0, stream>>>(Dm, knnd, knni, N);
}